// PointNetSetAbstraction_35416300323428
// MI455X (gfx1250) — compile-verified
//
#include <hip/hip_runtime.h>
#include <hip/hip_bf16.h>
#include <float.h>

typedef __attribute__((ext_vector_type(2))) float v2f;
typedef __attribute__((ext_vector_type(8))) float v8f;

#define B_ 8
#define N_ 4096
#define D_ 64
#define K_ 16
#define M_ 65536          /* N_*K_ columns per batch */
#define EPS_ 1e-5f
#define SLOPE_ 0.1f

/* ---- workspace layout (units: 4-byte slots). Requires ~548 MB of ws. ---- */
#define WS_IDX    0                         /* B*N*K ints                    */
#define WS_STATS0 (B_*N_*K_)                /* B*64*2 floats (sum, sumsq)    */
#define WS_STATS1 (WS_STATS0 + B_*64*2)
#define WS_STATS2 (WS_STATS1 + B_*64*2)
#define WS_BUFA   (WS_STATS2 + B_*128*2)    /* X0 (68 ch), later reused = Y1 */
#define WS_BUFB   (WS_BUFA + (size_t)B_*68*M_)   /* Y0 (64 ch) */
#define WS_BUFC   (WS_BUFB + (size_t)B_*64*M_)   /* Y2 (128 ch) */

/* ---------------- KNN: fused distance + top-16 selection ---------------- */
__global__ void pn_knn_kernel(const float* __restrict__ xyz,
                              int* __restrict__ idx_ws,
                              float* __restrict__ out_xyz,
                              int* __restrict__ out_idx) {
  __shared__ float sx[N_], sy[N_], sz[N_];          /* 48 KB LDS */
  const int b    = blockIdx.x >> 4;                 /* N_/256 = 16 blocks/batch */
  const int tile = blockIdx.x & 15;
  const float* base = xyz + (size_t)b * 3 * N_;
  for (int j = threadIdx.x; j < N_; j += 256) {
    sx[j] = base[j];
    sy[j] = base[N_ + j];
    sz[j] = base[2 * N_ + j];
  }
  __syncthreads();

  const int q = tile * 256 + threadIdx.x;
  const float qx = sx[q], qy = sy[q], qz = sz[q];

  float bd[K_]; int bi[K_];
#pragma unroll
  for (int s = 0; s < K_; ++s) { bd[s] = FLT_MAX; bi[s] = q; }
  float worst = FLT_MAX; int wslot = 0;

  for (int j = 0; j < N_; ++j) {
    const float dx = sx[j] - qx, dy = sy[j] - qy, dz = sz[j] - qz;
    const float d = dx * dx + dy * dy + dz * dz;
    if (d < worst) {
#pragma unroll
      for (int s = 0; s < K_; ++s) if (s == wslot) { bd[s] = d; bi[s] = j; }
      worst = bd[0]; wslot = 0;
#pragma unroll
      for (int s = 1; s < K_; ++s) if (bd[s] > worst) { worst = bd[s]; wslot = s; }
    }
  }

  const size_t off = ((size_t)b * N_ + q) * K_;
#pragma unroll
  for (int s = 0; s < K_; ++s) { idx_ws[off + s] = bi[s]; out_idx[off + s] = bi[s]; }

  /* xyz passthrough output */
  out_xyz[(size_t)b * 3 * N_ + q]            = qx;
  out_xyz[(size_t)b * 3 * N_ + N_ + q]       = qy;
  out_xyz[(size_t)b * 3 * N_ + 2 * N_ + q]   = qz;
}

/* --------- gather grouped features into padded X0 [B][68][M_] ----------- */
__global__ void pn_gather_kernel(const float* __restrict__ xyz,
                                 const float* __restrict__ pts,
                                 const int* __restrict__ idx_ws,
                                 float* __restrict__ X0) {
  const int t   = blockIdx.x * 256 + threadIdx.x;   /* B*N*K threads */
  const int b   = t / (N_ * K_);
  const int rem = t % (N_ * K_);                    /* = m = n*K + k */
  const int n   = rem / K_;
  const int m   = rem;
  const int j   = idx_ws[(size_t)t];

  const float* xb = xyz + (size_t)b * 3 * N_;
  const float cx = xb[n], cy = xb[N_ + n], cz = xb[2 * N_ + n];

  float* Xb = X0 + (size_t)b * 68 * M_;
  Xb[0 * (size_t)M_ + m] = xb[j] - cx;
  Xb[1 * (size_t)M_ + m] = xb[N_ + j] - cy;
  Xb[2 * (size_t)M_ + m] = xb[2 * N_ + j] - cz;

  const float* pb = pts + (size_t)b * D_ * N_;
#pragma unroll 4
  for (int c = 0; c < D_; ++c)
    Xb[(size_t)(3 + c) * M_ + m] = pb[(size_t)c * N_ + j];
  Xb[67 * (size_t)M_ + m] = 0.0f;                   /* zero pad channel */
}

/* --- GEMM via V_WMMA_F32_16X16X4_F32 + fused per-channel sum/sumsq ------ */
/* Y[b][co][m] = sum_ci W[co][ci] * X[b][ci][m]; wave owns 16 co x 256 m.  */
template <int STEPS>
__global__ void pn_gemm_stats_kernel(const float* __restrict__ X,
                                     const float* __restrict__ W,
                                     float* __restrict__ Y,
                                     float* __restrict__ stats,
                                     int cin, int cout, int cinPad) {
  const int lane = threadIdx.x & 31;
  const int wave = threadIdx.x >> 5;
  const int mGroups = M_ / 256;                     /* 256 */
  const int wavesPerBatch = (cout >> 4) * mGroups;

  const int gw = blockIdx.x * 8 + wave;
  const int b  = gw / wavesPerBatch;
  const int r  = gw % wavesPerBatch;
  const int coT = r / mGroups;
  const int mg  = r % mGroups;
  const int mbase = mg * 256;
  const int co0 = coT * 16;

  /* A fragment layout (32-bit A 16x4): v0 = K=0|K=2, v1 = K=1|K=3 */
  const int arow = co0 + (lane & 15);
  const int koff = (lane < 16) ? 0 : 2;

  v2f a[STEPS];
#pragma unroll
  for (int s = 0; s < STEPS; ++s) {
    const int ci = s * 4 + koff;
    a[s].x = (ci     < cin) ? W[(size_t)arow * cin + ci]     : 0.0f;
    a[s].y = (ci + 1 < cin) ? W[(size_t)arow * cin + ci + 1] : 0.0f;
  }

  const float* Xb = X + (size_t)b * cinPad * M_;
  float* Yb = Y + (size_t)b * cout * M_;

  float sum[8], ssq[8];
#pragma unroll
  for (int r2 = 0; r2 < 8; ++r2) { sum[r2] = 0.f; ssq[r2] = 0.f; }

  for (int t = 0; t < 16; ++t) {
    const int m = mbase + t * 16 + (lane & 15);
    v8f acc = {0.f, 0.f, 0.f, 0.f, 0.f, 0.f, 0.f, 0.f};
#pragma unroll
    for (int s = 0; s < STEPS; ++s) {
      const int ci = s * 4 + koff;
      v2f bf;
      bf.x = Xb[(size_t)ci * M_ + m];
      bf.y = Xb[(size_t)(ci + 1) * M_ + m];
      acc = __builtin_amdgcn_wmma_f32_16x16x4_f32(false, a[s], false, bf,
                                                  (short)0, acc, false, false);
    }
    const int rowAdd = (lane < 16) ? 0 : 8;   /* D layout: vgpr r = rows r / r+8 */
#pragma unroll
    for (int r2 = 0; r2 < 8; ++r2) {
      const float v = acc[r2];
      Yb[(size_t)(co0 + r2 + rowAdd) * M_ + m] = v;
      sum[r2] += v; ssq[r2] += v * v;
    }
  }

  /* reduce across the 16 lanes that share a channel, then one atomic each */
#pragma unroll
  for (int r2 = 0; r2 < 8; ++r2) {
    float s = sum[r2], q = ssq[r2];
    for (int o = 1; o < 16; o <<= 1) {
      s += __shfl_xor(s, o, 32);
      q += __shfl_xor(q, o, 32);
    }
    if ((lane & 15) == 0) {
      const int ch = co0 + r2 + ((lane >> 4) << 3);
      atomicAdd(&stats[((size_t)b * cout + ch) * 2 + 0], s);
      atomicAdd(&stats[((size_t)b * cout + ch) * 2 + 1], q);
    }
  }
}

/* --------------- InstanceNorm(affine) + LeakyReLU, in place ------------- */
__global__ void pn_norm_lrelu_kernel(float* __restrict__ Y,
                                     const float* __restrict__ stats,
                                     const float* __restrict__ g,
                                     const float* __restrict__ bta, int cout) {
  const size_t i = (size_t)blockIdx.x * 256 + threadIdx.x;
  const int c = (int)((i / M_) % cout);
  const int b = (int)(i / ((size_t)M_ * cout));
  const float s = stats[((size_t)b * cout + c) * 2 + 0];
  const float q = stats[((size_t)b * cout + c) * 2 + 1];
  const float invM = 1.0f / (float)M_;
  const float mu = s * invM;
  const float var = fmaxf(q * invM - mu * mu, 0.0f);
  float x = Y[i];
  x = (x - mu) * rsqrtf(var + EPS_) * g[c] + bta[c];
  Y[i] = (x >= 0.0f) ? x : SLOPE_ * x;
}

/* --------- last layer: norm + LeakyReLU + max over K, write out --------- */
__global__ void pn_norm_maxpool_kernel(const float* __restrict__ Y2,
                                       const float* __restrict__ stats,
                                       const float* __restrict__ g,
                                       const float* __restrict__ bta,
                                       float* __restrict__ out_np) {
  const int t = blockIdx.x * 256 + threadIdx.x;   /* B*128*N threads */
  const int n = t % N_;
  const int c = (t / N_) % 128;
  const int b = t / (N_ * 128);
  const float s = stats[((size_t)b * 128 + c) * 2 + 0];
  const float q = stats[((size_t)b * 128 + c) * 2 + 1];
  const float invM = 1.0f / (float)M_;
  const float mu = s * invM;
  const float rstd = rsqrtf(fmaxf(q * invM - mu * mu, 0.0f) + EPS_);
  const float gg = g[c], bb = bta[c];
  const float* src = Y2 + ((size_t)b * 128 + c) * M_ + (size_t)n * K_;
  float best = -FLT_MAX;
#pragma unroll
  for (int k = 0; k < K_; ++k) {
    float x = (src[k] - mu) * rstd * gg + bb;
    x = (x >= 0.0f) ? x : SLOPE_ * x;
    best = fmaxf(best, x);
  }
  out_np[t] = best;                               /* t == ((b*128+c)*N + n) */
}

extern "C" void kernel_launch(void* const* d_in, const int* in_sizes, int n_in,
                              void* d_out, int out_size, void* d_ws, size_t ws_size,
                              hipStream_t stream) {
  (void)in_sizes; (void)n_in; (void)out_size; (void)ws_size;
  const float* xyz = (const float*)d_in[0];
  const float* pts = (const float*)d_in[1];
  const float* W0  = (const float*)d_in[2];
  const float* W1  = (const float*)d_in[3];
  const float* W2  = (const float*)d_in[4];
  const float* g0  = (const float*)d_in[5];
  const float* b0  = (const float*)d_in[6];
  const float* g1  = (const float*)d_in[7];
  const float* b1  = (const float*)d_in[8];
  const float* g2  = (const float*)d_in[9];
  const float* b2  = (const float*)d_in[10];

  float* ws = (float*)d_ws;
  int*   idx_ws = (int*)d_ws;
  float* stats0 = ws + WS_STATS0;
  float* stats1 = ws + WS_STATS1;
  float* stats2 = ws + WS_STATS2;
  float* bufA = ws + WS_BUFA;   /* X0, later Y1 */
  float* bufB = ws + WS_BUFB;   /* Y0 */
  float* bufC = ws + WS_BUFC;   /* Y2 */

  float* out     = (float*)d_out;
  float* out_xyz = out;                                   /* B*3*N     */
  float* out_np  = out + (size_t)B_ * 3 * N_;             /* B*128*N   */
  int*   out_idx = (int*)(out + (size_t)B_ * 3 * N_ + (size_t)B_ * 128 * N_);

  hipMemsetAsync(stats0, 0,
                 (size_t)(B_ * 64 * 2 + B_ * 64 * 2 + B_ * 128 * 2) * sizeof(float),
                 stream);

  pn_knn_kernel<<<B_ * (N_ / 256), 256, 0, stream>>>(xyz, idx_ws, out_xyz, out_idx);
  pn_gather_kernel<<<(B_ * N_ * K_) / 256, 256, 0, stream>>>(xyz, pts, idx_ws, bufA);

  /* layer 1: Cin=67 (pad 68) -> Cout=64 */
  {
    const int cout = 64;
    const int waves = B_ * (cout / 16) * (M_ / 256);
    pn_gemm_stats_kernel<17><<<waves / 8, 256, 0, stream>>>(bufA, W0, bufB, stats0, 67, cout, 68);
    pn_norm_lrelu_kernel<<<(int)(((size_t)B_ * cout * M_) / 256), 256, 0, stream>>>(bufB, stats0, g0, b0, cout);
  }
  /* layer 2: 64 -> 64 (X0 region reused for Y1) */
  {
    const int cout = 64;
    const int waves = B_ * (cout / 16) * (M_ / 256);
    pn_gemm_stats_kernel<16><<<waves / 8, 256, 0, stream>>>(bufB, W1, bufA, stats1, 64, cout, 64);
    pn_norm_lrelu_kernel<<<(int)(((size_t)B_ * cout * M_) / 256), 256, 0, stream>>>(bufA, stats1, g1, b1, cout);
  }
  /* layer 3: 64 -> 128 */
  {
    const int cout = 128;
    const int waves = B_ * (cout / 16) * (M_ / 256);
    pn_gemm_stats_kernel<16><<<waves / 8, 256, 0, stream>>>(bufA, W2, bufC, stats2, 64, cout, 64);
  }
  /* fused norm + LeakyReLU + max over K */
  pn_norm_maxpool_kernel<<<(B_ * 128 * N_) / 256, 256, 0, stream>>>(bufC, stats2, g2, b2, out_np);
}